// SSMMemory_64424509440808
// MI455X (gfx1250) — compile-verified
//
#include <hip/hip_runtime.h>
#include <hip/hip_bf16.h>
#include <math.h>

// Problem constants (from reference)
constexpr int Lc      = 24;
constexpr int Bc      = 64;
constexpr int Dc      = 768;
constexpr int Ec      = 1536;
constexpr int DCONVc  = 4;
constexpr int DSTATEc = 16;
constexpr int DTRANKc = 48;
constexpr int XPNc    = DTRANKc + 2 * DSTATEc; // 80

typedef __attribute__((ext_vector_type(2))) float v2f;
typedef __attribute__((ext_vector_type(8))) float v8f;
typedef __attribute__((ext_vector_type(4))) int   v4i;

// Address-space-qualified int4 pointers matching the async builtin signature:
//   __builtin_amdgcn_global_load_async_to_lds_b128(int4 AS1*, int4 AS3*, i32, i32)
typedef __attribute__((address_space(1))) v4i gv4i;
typedef __attribute__((address_space(3))) v4i lv4i;

#if defined(__has_builtin)
#if __has_builtin(__builtin_amdgcn_global_load_async_to_lds_b128)
#define HAVE_ASYNC_LDS 1
#endif
#if __has_builtin(__builtin_amdgcn_s_wait_asynccnt)
#define HAVE_WAIT_ASYNC 1
#endif
#endif

// Stage 16 bytes global -> LDS. Async (ASYNCcnt-tracked DMA) when available.
__device__ __forceinline__ void stage_b128(const float* g, float* l) {
#ifdef HAVE_ASYNC_LDS
    __builtin_amdgcn_global_load_async_to_lds_b128((gv4i*)g, (lv4i*)l, 0, 0);
#else
    *(float4*)l = *(const float4*)g;
#endif
}

template<int N>
__device__ __forceinline__ void wait_async() {
#ifdef HAVE_ASYNC_LDS
#ifdef HAVE_WAIT_ASYNC
    __builtin_amdgcn_s_wait_asynccnt(N);
#else
    asm volatile("s_wait_asynccnt %0" :: "i"(N) : "memory");
#endif
#endif
}

// ---------------------------------------------------------------------------
// RMSNorm: one block per row, LDS tree reduction.
// ---------------------------------------------------------------------------
__global__ __launch_bounds__(256)
void rmsnorm_kernel(const float* __restrict__ x, const float* __restrict__ w,
                    float* __restrict__ out, int n) {
    __shared__ float red[256];
    const int b = blockIdx.x;
    const int t = threadIdx.x;
    const float* xr = x + (long)b * n;
    float s = 0.f;
    for (int i = t; i < n; i += 256) { float v = xr[i]; s += v * v; }
    red[t] = s;
    __syncthreads();
    for (int off = 128; off > 0; off >>= 1) {
        if (t < off) red[t] += red[t + off];
        __syncthreads();
    }
    const float scale = rsqrtf(red[0] / (float)n + 1e-5f);
    for (int i = t; i < n; i += 256) out[(long)b * n + i] = xr[i] * scale * w[i];
}

// ---------------------------------------------------------------------------
// Direct-global M=64 WMMA GEMM (small operands): C[64,N] = A[64,K]*W[N,K]^T
//   grid.x = N/16, block = 128 (wave w owns M rows [16w,16w+16))
// EPI: 0 = none, 1 = softplus(v + bias[col])
// ---------------------------------------------------------------------------
template<int EPI>
__global__ __launch_bounds__(128)
void gemm64_kernel(const float* __restrict__ A, const float* __restrict__ W,
                   float* __restrict__ Cout, int K, int lda, int ldw, int ldc,
                   const float* __restrict__ bias) {
    const int lane = threadIdx.x & 31;
    const int wave = threadIdx.x >> 5;
    const int half = lane >> 4;
    const int l16  = lane & 15;
    const int m0   = wave * 16;
    const int n0   = blockIdx.x * 16;

    const float* __restrict__ arow = A + (long)(m0 + l16) * lda;
    const float* __restrict__ wrow = W + (long)(n0 + l16) * ldw;
    const int koff = half * 2;

    v8f acc = {};
    for (int k0 = 0; k0 < K; k0 += 4) {
        v2f a, b;
        a.x = arow[k0 + koff];
        a.y = arow[k0 + koff + 1];
        b.x = wrow[k0 + koff];
        b.y = wrow[k0 + koff + 1];
        __builtin_prefetch(&wrow[k0 + 256], 0, 1);
        acc = __builtin_amdgcn_wmma_f32_16x16x4_f32(
            false, a, false, b, (short)0, acc, false, false);
    }

    const int col = n0 + l16;
#pragma unroll
    for (int r = 0; r < 8; ++r) {
        const int row = m0 + r + half * 8;
        float v = acc[r];
        if (EPI == 1) {
            v += bias[col];
            v = (v > 20.f) ? v : log1pf(__expf(v));   // softplus
        }
        Cout[(long)row * ldc + col] = v;
    }
}

// ---------------------------------------------------------------------------
// LDS-staged M=64 WMMA GEMM for the big weight streams (K % 32 == 0).
// Block = 128 threads (4 waves). The 16(N) x 32(K) weight chunk is DMA'd to
// LDS once per block (global_load_async_to_lds_b128, double-buffered,
// s_wait_asynccnt + barrier) and shared by all 4 M-tiles -> 4x fewer weight
// load issues on the dominant HBM stream. A (activations, L2-resident) is
// read directly from global.
// EPI: 0 = none, 2 = v + resid[row,col]
// ---------------------------------------------------------------------------
template<int EPI>
__global__ __launch_bounds__(128)
void gemm64_lds_kernel(const float* __restrict__ A, const float* __restrict__ W,
                       float* __restrict__ Cout, int K, int lda, int ldw, int ldc,
                       const float* __restrict__ resid) {
    constexpr int KC      = 32;  // k-chunk (8 wmma steps)
    constexpr int LSTRIDE = 36;  // padded row stride in floats (144B: 16B-aligned)
    __shared__ float bsm[2][16 * LSTRIDE];

    const int t    = threadIdx.x;
    const int lane = t & 31;
    const int wave = t >> 5;
    const int half = lane >> 4;
    const int l16  = lane & 15;
    const int m0   = wave * 16;
    const int n0   = blockIdx.x * 16;
    const int koff = half * 2;

    // Loader mapping: 128 threads = 16 rows x 8 sixteen-byte segments.
    const int lrow = t >> 3;
    const int lseg = t & 7;
    const float* wsrc = W + (long)(n0 + lrow) * ldw + lseg * 4;
    float* ldst[2] = { &bsm[0][lrow * LSTRIDE + lseg * 4],
                       &bsm[1][lrow * LSTRIDE + lseg * 4] };

    const float* arow = A + (long)(m0 + l16) * lda;
    const int nch = K / KC;

    stage_b128(wsrc, ldst[0]);   // prime chunk 0

    v8f acc = {};
    for (int c = 0; c < nch; ++c) {
        if (c + 1 < nch) {
            stage_b128(wsrc + (long)(c + 1) * KC, ldst[(c + 1) & 1]);
            wait_async<1>();     // newest may fly; chunk c is complete
        } else {
            wait_async<0>();
        }
        __syncthreads();

        const float* abase = arow + c * KC;
        const float* bbuf  = &bsm[c & 1][l16 * LSTRIDE];
#pragma unroll
        for (int kk = 0; kk < KC; kk += 4) {
            v2f a, b;
            a.x = abase[kk + koff];
            a.y = abase[kk + koff + 1];
            b.x = bbuf[kk + koff];
            b.y = bbuf[kk + koff + 1];
            acc = __builtin_amdgcn_wmma_f32_16x16x4_f32(
                false, a, false, b, (short)0, acc, false, false);
        }
        __syncthreads();         // all waves done with buf (c&1) before reuse
    }

    const int col = n0 + l16;
#pragma unroll
    for (int r = 0; r < 8; ++r) {
        const int row = m0 + r + half * 8;
        float v = acc[r];
        if (EPI == 2) v += resid[(long)row * ldc + col];
        Cout[(long)row * ldc + col] = v;
    }
}

// ---------------------------------------------------------------------------
// Depthwise conv (DCONV=4, state shift) + SiLU -> xc[B,E]
// ---------------------------------------------------------------------------
__global__ __launch_bounds__(256)
void conv_silu_kernel(const float* __restrict__ xz,   // [B, 2E], xi first E cols
                      const float* __restrict__ cs,   // [B, E, DCONV]
                      const float* __restrict__ cw,   // [E, DCONV]
                      const float* __restrict__ cb,   // [E]
                      float* __restrict__ xc) {       // [B, E]
    const int idx = blockIdx.x * blockDim.x + threadIdx.x;  // b*E + e
    const int b = idx / Ec;
    const int e = idx - b * Ec;
    const float* csp = cs + ((long)b * Ec + e) * DCONVc;
    const float* cwp = cw + (long)e * DCONVc;
    const float xi = xz[(long)b * (2 * Ec) + e];
    float v = cb[e]
            + csp[1] * cwp[0]
            + csp[2] * cwp[1]
            + csp[3] * cwp[2]
            + xi     * cwp[3];
    xc[idx] = v / (1.f + __expf(-v));                 // silu
}

// ---------------------------------------------------------------------------
// SSM state update + readout + z-gate -> y[B,E]
// ---------------------------------------------------------------------------
__global__ __launch_bounds__(256)
void ssm_kernel(const float* __restrict__ dtf,   // [B,E]
                const float* __restrict__ xc,    // [B,E]
                const float* __restrict__ xz,    // [B,2E], z in second E cols
                const float* __restrict__ xdb,   // [B,80]
                const float* __restrict__ alog,  // [E,DSTATE]
                const float* __restrict__ dpar,  // [E]
                const float* __restrict__ ss,    // [B,E,DSTATE]
                float* __restrict__ y) {         // [B,E]
    const int idx = blockIdx.x * blockDim.x + threadIdx.x;  // b*E + e
    const int b = idx / Ec;
    const int e = idx - b * Ec;
    const float dt  = dtf[idx];
    const float xcv = xc[idx];
    const float* al  = alog + (long)e * DSTATEc;
    const float* ssp = ss + ((long)b * Ec + e) * DSTATEc;
    const float* Bm  = xdb + (long)b * XPNc + DTRANKc;
    const float* Cm  = Bm + DSTATEc;
    float acc = 0.f;
#pragma unroll
    for (int n = 0; n < DSTATEc; ++n) {
        const float Aneg = -__expf(al[n]);
        const float dA   = __expf(dt * Aneg);
        const float sn   = ssp[n] * dA + xcv * dt * Bm[n];
        acc += sn * Cm[n];
    }
    acc += dpar[e] * xcv;
    const float z = xz[(long)b * (2 * Ec) + Ec + e];
    y[idx] = acc * (z / (1.f + __expf(-z)));
}

// ---------------------------------------------------------------------------
// Host-side launch: 24 sequential layers, then final rmsnorm.
// ---------------------------------------------------------------------------
extern "C" void kernel_launch(void* const* d_in, const int* in_sizes, int n_in,
                              void* d_out, int out_size, void* d_ws, size_t ws_size,
                              hipStream_t stream) {
    const float* x_in      = (const float*)d_in[0];   // [B, D]
    const float* norm_w    = (const float*)d_in[1];   // [L, D]
    const float* in_w      = (const float*)d_in[2];   // [L, 2E, D]
    const float* conv_w    = (const float*)d_in[3];   // [L, E, DCONV]
    const float* conv_b    = (const float*)d_in[4];   // [L, E]
    const float* xp_w      = (const float*)d_in[5];   // [L, 80, E]
    const float* dt_w      = (const float*)d_in[6];   // [L, E, DTRANK]
    const float* dt_b      = (const float*)d_in[7];   // [L, E]
    const float* A_log     = (const float*)d_in[8];   // [L, E, DSTATE]
    const float* D_param   = (const float*)d_in[9];   // [L, E]
    const float* out_w     = (const float*)d_in[10];  // [L, D, E]
    const float* norm_f_w  = (const float*)d_in[11];  // [D]
    const float* conv_st   = (const float*)d_in[12];  // [L, B, E, DCONV]
    const float* ssm_st    = (const float*)d_in[13];  // [L, B, E, DSTATE]
    float* out = (float*)d_out;                       // [B, D]

    // Workspace carve-up (floats)
    float* ws   = (float*)d_ws;
    float* xA   = ws;                 // [B, D]   carry x (ping)
    float* xB   = xA  + Bc * Dc;      // [B, D]   carry x (pong)
    float* hbuf = xB  + Bc * Dc;      // [B, D]   rmsnorm output
    float* xz   = hbuf+ Bc * Dc;      // [B, 2E]
    float* xc   = xz  + Bc * 2 * Ec;  // [B, E]
    float* xdb  = xc  + Bc * Ec;      // [B, 80]
    float* dtf  = xdb + Bc * XPNc;    // [B, E]
    float* ybuf = dtf + Bc * Ec;      // [B, E]

    (void)hipMemcpyAsync(xA, x_in, (size_t)Bc * Dc * sizeof(float),
                         hipMemcpyDeviceToDevice, stream);

    float* cur = xA;
    float* nxt = xB;

    for (int l = 0; l < Lc; ++l) {
        const float* nw  = norm_w  + (long)l * Dc;
        const float* iw  = in_w    + (long)l * 2 * Ec * Dc;
        const float* cw  = conv_w  + (long)l * Ec * DCONVc;
        const float* cb  = conv_b  + (long)l * Ec;
        const float* xpw = xp_w    + (long)l * XPNc * Ec;
        const float* dtw = dt_w    + (long)l * Ec * DTRANKc;
        const float* dtb = dt_b    + (long)l * Ec;
        const float* al  = A_log   + (long)l * Ec * DSTATEc;
        const float* dp  = D_param + (long)l * Ec;
        const float* ow  = out_w   + (long)l * Dc * Ec;
        const float* cs  = conv_st + (long)l * Bc * Ec * DCONVc;
        const float* ss  = ssm_st  + (long)l * Bc * Ec * DSTATEc;

        // h = rmsnorm(x, norm_w[l])
        rmsnorm_kernel<<<Bc, 256, 0, stream>>>(cur, nw, hbuf, Dc);

        // xz = h @ in_w[l]^T   (M=64, N=3072, K=768)  [LDS-staged weights]
        gemm64_lds_kernel<0><<<2 * Ec / 16, 128, 0, stream>>>(
            hbuf, iw, xz, Dc, Dc, Dc, 2 * Ec, nullptr);

        // xc = silu(conv(cs<<1 | xi) + cb)
        conv_silu_kernel<<<(Bc * Ec) / 256, 256, 0, stream>>>(xz, cs, cw, cb, xc);

        // xdb = xc @ xp_w[l]^T  (M=64, N=80, K=1536)
        gemm64_kernel<0><<<XPNc / 16, 128, 0, stream>>>(
            xc, xpw, xdb, Ec, Ec, Ec, XPNc, nullptr);

        // dtf = softplus(xdb[:, :48] @ dt_w[l]^T + dt_b[l])  (M=64, N=1536, K=48)
        gemm64_kernel<1><<<Ec / 16, 128, 0, stream>>>(
            xdb, dtw, dtf, DTRANKc, XPNc, DTRANKc, Ec, dtb);

        // y = (ss*dA + xc*dt*Bm).Cm + D*xc, gated by silu(z)
        ssm_kernel<<<(Bc * Ec) / 256, 256, 0, stream>>>(
            dtf, xc, xz, xdb, al, dp, ss, ybuf);

        // x_next = x + y @ out_w[l]^T  (M=64, N=768, K=1536)  [LDS-staged]
        gemm64_lds_kernel<2><<<Dc / 16, 128, 0, stream>>>(
            ybuf, ow, nxt, Ec, Ec, Ec, Dc, cur);

        float* t = cur; cur = nxt; nxt = t;
    }

    // final rmsnorm with norm_f_w -> d_out
    rmsnorm_kernel<<<Bc, 256, 0, stream>>>(cur, norm_f_w, out, Dc);
}